// CrossNetwork_18262200943284
// MI455X (gfx1250) — compile-verified
//
#include <hip/hip_runtime.h>
#include <stdint.h>

#ifndef __has_builtin
#define __has_builtin(x) 0
#endif

// Problem geometry (fixed by the reference): B=16384 rows, D=1024, L=4 layers.
#define DCN_D 1024
#define DCN_L 4
#define CHUNKS 8              // 8 x float4 per lane * 32 lanes * 4 = 1024 = D
#define WAVES_PER_BLOCK 8
#define BLOCK_THREADS (WAVES_PER_BLOCK * 32)

typedef float fvec4 __attribute__((ext_vector_type(4)));
typedef int   v4i   __attribute__((vector_size(16)));   // matches builtin's V4i

// Pointer casts matching the async-builtin signature:
//   arg0: int4 addrspace(1)* (global), arg1: int4 addrspace(3)* (LDS)
#define AS1P(p) ((__attribute__((address_space(1))) v4i*)(p))
#define AS3P(p) ((__attribute__((address_space(3))) v4i*)(p))

__global__ __launch_bounds__(BLOCK_THREADS) void
dcn_cross4_kernel(const float* __restrict__ x,
                  const float* __restrict__ w,
                  const float* __restrict__ b,
                  float* __restrict__ out,
                  int nrows)
{
    // Stage the broadcast weight/bias tables (4*1024 fp32 each = 16 KB + 16 KB)
    // into LDS once per block.
    __shared__ float s_w[DCN_L * DCN_D];
    __shared__ float s_b[DCN_L * DCN_D];

    const int tid = threadIdx.x;

#if __has_builtin(__builtin_amdgcn_global_load_async_to_lds_b128) && \
    __has_builtin(__builtin_amdgcn_s_wait_asynccnt)
    // CDNA5 async DMA path: GLOBAL_LOAD_ASYNC_TO_LDS_B128, tracked by ASYNCcnt.
    #pragma unroll
    for (int i = 0; i < (DCN_L * DCN_D) / (BLOCK_THREADS * 4); ++i) {
        const int fidx = (i * BLOCK_THREADS + tid) * 4;   // float index, 16B granules
        __builtin_amdgcn_global_load_async_to_lds_b128(AS1P(w + fidx), AS3P(&s_w[fidx]), 0, 0);
        __builtin_amdgcn_global_load_async_to_lds_b128(AS1P(b + fidx), AS3P(&s_b[fidx]), 0, 0);
    }
    __builtin_amdgcn_s_wait_asynccnt(0);
#else
    // Fallback: plain vector copy through VGPRs.
    #pragma unroll
    for (int i = 0; i < (DCN_L * DCN_D) / (BLOCK_THREADS * 4); ++i) {
        const int fidx = (i * BLOCK_THREADS + tid) * 4;
        *(fvec4*)&s_w[fidx] = *(const fvec4*)&w[fidx];
        *(fvec4*)&s_b[fidx] = *(const fvec4*)&b[fidx];
    }
#endif
    __syncthreads();

    // One wave32 per row; each lane owns 32 columns as 8 interleaved float4s
    // (lane-contiguous b128 accesses -> fully coalesced 512B per instruction).
    const int  wave = tid >> 5;
    const int  lane = tid & 31;
    const long row  = (long)blockIdx.x * WAVES_PER_BLOCK + wave;
    if (row >= nrows) return;

    const fvec4* __restrict__ xrow = (const fvec4*)(x + row * DCN_D);

    fvec4 x0[CHUNKS], xc[CHUNKS];
    #pragma unroll
    for (int c = 0; c < CHUNKS; ++c) {
        // Streamed exactly once: non-temporal load (TH=NT), keep in registers.
        x0[c] = __builtin_nontemporal_load(&xrow[c * 32 + lane]);
        xc[c] = x0[c];
    }

    #pragma unroll
    for (int l = 0; l < DCN_L; ++l) {
        const fvec4* wl = (const fvec4*)(s_w) + l * (DCN_D / 4);
        const fvec4* bl = (const fvec4*)(s_b) + l * (DCN_D / 4);

        // Per-lane partial dot over this lane's 32 columns.
        float p = 0.0f;
        #pragma unroll
        for (int c = 0; c < CHUNKS; ++c) {
            const fvec4 wv = wl[c * 32 + lane];   // ds_load_b128
            const fvec4 xv = xc[c];
            p = __builtin_fmaf(xv.x, wv.x, p);
            p = __builtin_fmaf(xv.y, wv.y, p);
            p = __builtin_fmaf(xv.z, wv.z, p);
            p = __builtin_fmaf(xv.w, wv.w, p);
        }

        // wave32 butterfly reduction -> full dot s broadcast to all lanes.
        #pragma unroll
        for (int m = 16; m >= 1; m >>= 1)
            p += __shfl_xor(p, m, 32);

        // x_{l+1} = x0 * s + b_l + x_l   (elementwise, fp32)
        #pragma unroll
        for (int c = 0; c < CHUNKS; ++c) {
            const fvec4 bv = bl[c * 32 + lane];   // ds_load_b128
            xc[c] = x0[c] * p + bv + xc[c];
        }
    }

    fvec4* __restrict__ orow = (fvec4*)(out + row * DCN_D);
    #pragma unroll
    for (int c = 0; c < CHUNKS; ++c)
        __builtin_nontemporal_store(xc[c], &orow[c * 32 + lane]);
}

extern "C" void kernel_launch(void* const* d_in, const int* in_sizes, int n_in,
                              void* d_out, int out_size, void* d_ws, size_t ws_size,
                              hipStream_t stream) {
    (void)n_in; (void)out_size; (void)d_ws; (void)ws_size;
    const float* x = (const float*)d_in[0];   // (B, D) fp32
    const float* w = (const float*)d_in[1];   // (L, D) fp32
    const float* b = (const float*)d_in[2];   // (L, D) fp32
    float* out = (float*)d_out;               // (B, D) fp32

    const int nrows = in_sizes[0] / DCN_D;    // B = 16384
    const int grid  = (nrows + WAVES_PER_BLOCK - 1) / WAVES_PER_BLOCK;

    dcn_cross4_kernel<<<grid, BLOCK_THREADS, 0, stream>>>(x, w, b, out, nrows);
}